// AttentionBlock_36971078484521
// MI455X (gfx1250) — compile-verified
//
#include <hip/hip_runtime.h>
#include <hip/hip_bf16.h>

// ---------------------------------------------------------------------------
// AttentionBlock for MI455X (gfx1250): fp32 WMMA (V_WMMA_F32_16X16X4_F32)
// throughout so results match the fp32 reference; flash-style attention with
// LDS staging; global_prefetch for next KV chunk.
//
// Workspace layout (floats), total 9,437,184 floats = 36 MB:
//   q_in  [8192*128]  kv_in [16384*128]
//   Qh    [8192*128]  Kh    [16384*128]  Vh [16384*128]
//   AO    [8192*128]
// ---------------------------------------------------------------------------

typedef float v2f __attribute__((ext_vector_type(2)));
typedef float v8f __attribute__((ext_vector_type(8)));

#define EMBED 128
#define HEADS 8
#define HD    16
#define BATCH 8
#define LQ    1024
#define LK    2048
#define NQ    (BATCH * LQ)   // 8192
#define NK    (BATCH * LK)   // 16384

__device__ __forceinline__ v8f wmma4(float a0, float a1, float b0, float b1, v8f c) {
  v2f a = {a0, a1};
  v2f b = {b0, b1};
  // D = A(16x4, f32) x B(4x16, f32) + C(16x16, f32)
  return __builtin_amdgcn_wmma_f32_16x16x4_f32(
      /*neg_a=*/false, a, /*neg_b=*/false, b,
      /*c_mod=*/(short)0, c, /*reuse_a=*/false, /*reuse_b=*/false);
}

// ---------------------------------------------------------------------------
// 1) Positional-table gather + add.  One 128-thread block per row.
// ---------------------------------------------------------------------------
__global__ __launch_bounds__(128)
void pos_add_kernel(const float* __restrict__ feat,
                    const int*   __restrict__ idx,
                    const float* __restrict__ table,
                    float* __restrict__ out) {
  const int row = blockIdx.x;
  const int t   = threadIdx.x;                 // 0..127
  const int* ip = idx + (size_t)row * 4;
  const int p   = ip[1] + ip[2] * 200 + ip[3] * 40000 + 1;   // <= 160000
  out[(size_t)row * EMBED + t] =
      feat[(size_t)row * EMBED + t] + table[(size_t)p * EMBED + t];
}

// ---------------------------------------------------------------------------
// 2) QKV projection: Y[n,o] = sum_e X[n,e] * W[o,e] + bias[o], then scatter
//    to head layout [B][H][LSEQ][HD] (scale folded in, 0.25 for Q).
//    Grid: (EMBED/16 col tiles, nrows/128).  Block: 256 threads = 8 waves,
//    wave w owns rows [blockIdx.y*128 + w*16, +16).  LSEQ is compile-time so
//    the batch split is shifts, not an integer-division expansion.
// ---------------------------------------------------------------------------
template <int LSEQ>
__global__ __launch_bounds__(256)
void qkv_proj_kernel(const float* __restrict__ X,     // [nrows,128]
                     const float* __restrict__ W,     // [128,128] sub-block
                     const float* __restrict__ bias,  // [128]
                     float* __restrict__ Yh,          // [B][H][LSEQ][HD]
                     float scale) {
  const int lane = threadIdx.x & 31;
  const int wave = threadIdx.x >> 5;
  const int col0 = blockIdx.x * 16;
  const int row0 = blockIdx.y * 128 + wave * 16;
  const int nfr  = lane & 15;              // A row (m) and B col (n)
  const int kb   = (lane >> 4) << 1;       // k-pair base: 0 or 2

  const float* xrow = X + (size_t)(row0 + nfr) * EMBED;
  const float* wrow = W + (size_t)(col0 + nfr) * EMBED;

  v8f acc = {0.f, 0.f, 0.f, 0.f, 0.f, 0.f, 0.f, 0.f};
#pragma unroll
  for (int k = 0; k < EMBED; k += 4) {
    const float a0 = xrow[k + kb];
    const float a1 = xrow[k + kb + 1];
    const float b0 = wrow[k + kb];
    const float b1 = wrow[k + kb + 1];
    acc = wmma4(a0, a1, b0, b1, acc);
  }

  const float bn = bias[col0 + nfr];
  // Entire 16-row tile lies in one batch (LSEQ is a multiple of 128).
  const int b  = row0 / LSEQ;              // compile-time LSEQ -> shift
  const int s0 = row0 % LSEQ;
  const int o  = col0 + nfr;
  const int h  = o >> 4;
  const int d  = o & 15;
  const int mh = (lane >> 4) << 3;         // +0 or +8 row offset
#pragma unroll
  for (int r = 0; r < 8; ++r) {
    const int s = s0 + r + mh;
    Yh[(((size_t)(b * HEADS + h) * LSEQ) + s) * HD + d] = (acc[r] + bn) * scale;
  }
}

// ---------------------------------------------------------------------------
// 3) Flash attention per (b,h): Q-tile = 128 rows per block, KV chunks of 64.
//    Grid: (LQ/128, B*HEADS).  Block: 256 threads = 8 waves; every wave owns
//    one 16-row m-tile for both S = Q*K^T and O += P*V (all waves busy in all
//    WMMA phases).  LDS ~57.5 KB -> 5 workgroups per 320 KB WGP.
// ---------------------------------------------------------------------------
__global__ __launch_bounds__(256)
void attn_kernel(const float* __restrict__ Qh,
                 const float* __restrict__ Kh,
                 const float* __restrict__ Vh,
                 float* __restrict__ AO) {          // [B][LQ][H][HD] = [8192][128]
  const int bh = blockIdx.y;
  const int b  = bh >> 3;
  const int h  = bh & 7;
  const int q0 = blockIdx.x * 128;

  const float* Qbase = Qh + ((size_t)bh * LQ + q0) * HD;
  const float* Kbase = Kh + (size_t)bh * LK * HD;
  const float* Vbase = Vh + (size_t)bh * LK * HD;

  __shared__ float sQ[128 * HD];    // 8 KB
  __shared__ float sK[64 * HD];     // 4 KB
  __shared__ float sV[64 * HD];     // 4 KB
  __shared__ float sS[128 * 64];    // 32 KB  (scores -> probabilities)
  __shared__ float sO[128 * HD];    // 8 KB   (running output)
  __shared__ float sMax[128], sSum[128], sAlpha[128];

  const int tid  = threadIdx.x;
  const int lane = tid & 31;
  const int wave = tid >> 5;        // == m-tile index (0..7)
  const int nfr  = lane & 15;
  const int kb   = (lane >> 4) << 1;
  const int mh   = (lane >> 4) << 3;

  for (int i = tid; i < 128 * HD; i += 256) { sQ[i] = Qbase[i]; sO[i] = 0.f; }
  if (tid < 128) { sMax[tid] = -3.0e38f; sSum[tid] = 0.f; }
  __syncthreads();

  for (int kv0 = 0; kv0 < LK; kv0 += 64) {
    // --- stage KV chunk into LDS -----------------------------------------
    for (int i = tid; i < 64 * HD; i += 256) {
      sK[i] = Kbase[(size_t)kv0 * HD + i];
      sV[i] = Vbase[(size_t)kv0 * HD + i];
    }
    if (kv0 + 64 < LK) {
      // gfx1250 global_prefetch_b8 of the next chunk's cachelines
      __builtin_prefetch(Kbase + (size_t)(kv0 + 64) * HD + tid * 4, 0, 1);
      __builtin_prefetch(Vbase + (size_t)(kv0 + 64) * HD + tid * 4, 0, 1);
    }
    __syncthreads();

    // --- S = Q * K^T: wave owns m-tile `wave`; A fragment reused across the
    //     4 n-tiles (4 live accumulators, 16 WMMA per wave per chunk) -------
    {
      v8f acc0 = {0.f, 0.f, 0.f, 0.f, 0.f, 0.f, 0.f, 0.f};
      v8f acc1 = acc0, acc2 = acc0, acc3 = acc0;
#pragma unroll
      for (int k = 0; k < HD; k += 4) {
        const float a0 = sQ[(wave * 16 + nfr) * HD + k + kb];
        const float a1 = sQ[(wave * 16 + nfr) * HD + k + kb + 1];
        // B[k][n] = K[n][k] per n-tile
        float b0, b1;
        b0 = sK[(0 * 16 + nfr) * HD + k + kb]; b1 = sK[(0 * 16 + nfr) * HD + k + kb + 1];
        acc0 = wmma4(a0, a1, b0, b1, acc0);
        b0 = sK[(1 * 16 + nfr) * HD + k + kb]; b1 = sK[(1 * 16 + nfr) * HD + k + kb + 1];
        acc1 = wmma4(a0, a1, b0, b1, acc1);
        b0 = sK[(2 * 16 + nfr) * HD + k + kb]; b1 = sK[(2 * 16 + nfr) * HD + k + kb + 1];
        acc2 = wmma4(a0, a1, b0, b1, acc2);
        b0 = sK[(3 * 16 + nfr) * HD + k + kb]; b1 = sK[(3 * 16 + nfr) * HD + k + kb + 1];
        acc3 = wmma4(a0, a1, b0, b1, acc3);
      }
#pragma unroll
      for (int r = 0; r < 8; ++r) {
        const int rr = (wave * 16 + r + mh) * 64;
        sS[rr + 0 * 16 + nfr] = acc0[r];
        sS[rr + 1 * 16 + nfr] = acc1[r];
        sS[rr + 2 * 16 + nfr] = acc2[r];
        sS[rr + 3 * 16 + nfr] = acc3[r];
      }
    }
    __syncthreads();

    // --- online softmax (one thread per query row, waves 0..3) -----------
    if (tid < 128) {
      float* row = sS + tid * 64;
      const float oldm = sMax[tid];
      float cm = row[0];
#pragma unroll 8
      for (int j = 1; j < 64; ++j) cm = fmaxf(cm, row[j]);
      const float newm  = fmaxf(oldm, cm);
      const float alpha = __expf(oldm - newm);
      float s = 0.f;
#pragma unroll 8
      for (int j = 0; j < 64; ++j) {
        const float e = __expf(row[j] - newm);
        row[j] = e;
        s += e;
      }
      sSum[tid]   = sSum[tid] * alpha + s;
      sMax[tid]   = newm;
      sAlpha[tid] = alpha;
    }
    __syncthreads();

    // --- rescale running O by alpha --------------------------------------
    for (int i = tid; i < 128 * HD; i += 256) sO[i] *= sAlpha[i >> 4];
    __syncthreads();

    // --- O += P * V: wave owns m-tile `wave`; K=64 -> 16 WMMA per wave ---
    {
      v8f acc;
#pragma unroll
      for (int r = 0; r < 8; ++r) acc[r] = sO[(wave * 16 + r + mh) * HD + nfr];
#pragma unroll
      for (int k = 0; k < 64; k += 4) {
        const float a0 = sS[(wave * 16 + nfr) * 64 + k + kb];
        const float a1 = sS[(wave * 16 + nfr) * 64 + k + kb + 1];
        const float b0 = sV[(k + kb) * HD + nfr];             // B[k][n]=V[k][n]
        const float b1 = sV[(k + kb + 1) * HD + nfr];
        acc = wmma4(a0, a1, b0, b1, acc);
      }
#pragma unroll
      for (int r = 0; r < 8; ++r) sO[(wave * 16 + r + mh) * HD + nfr] = acc[r];
    }
    __syncthreads();
  }

  // --- finalize: divide by row sums, scatter to [B][LQ][H][HD] -----------
  for (int i = tid; i < 128 * HD; i += 256) {
    const int m = i >> 4;
    const int d = i & 15;
    AO[((size_t)(b * LQ + q0 + m)) * EMBED + h * HD + d] = sO[i] / sSum[m];
  }
}

// ---------------------------------------------------------------------------
// 4) Output projection: Y = X @ out_w^T + out_b   (plain row-major store)
// ---------------------------------------------------------------------------
__global__ __launch_bounds__(256)
void out_proj_kernel(const float* __restrict__ X,     // [nrows,128]
                     const float* __restrict__ W,     // [128,128]
                     const float* __restrict__ bias,  // [128]
                     float* __restrict__ Y) {         // [nrows,128]
  const int lane = threadIdx.x & 31;
  const int wave = threadIdx.x >> 5;
  const int col0 = blockIdx.x * 16;
  const int row0 = blockIdx.y * 128 + wave * 16;
  const int nfr  = lane & 15;
  const int kb   = (lane >> 4) << 1;
  const int mh   = (lane >> 4) << 3;

  const float* xrow = X + (size_t)(row0 + nfr) * EMBED;
  const float* wrow = W + (size_t)(col0 + nfr) * EMBED;

  v8f acc = {0.f, 0.f, 0.f, 0.f, 0.f, 0.f, 0.f, 0.f};
#pragma unroll
  for (int k = 0; k < EMBED; k += 4) {
    const float a0 = xrow[k + kb];
    const float a1 = xrow[k + kb + 1];
    const float b0 = wrow[k + kb];
    const float b1 = wrow[k + kb + 1];
    acc = wmma4(a0, a1, b0, b1, acc);
  }

  const float bn = bias[col0 + nfr];
#pragma unroll
  for (int r = 0; r < 8; ++r)
    Y[(size_t)(row0 + r + mh) * EMBED + col0 + nfr] = acc[r] + bn;
}

// ---------------------------------------------------------------------------
extern "C" void kernel_launch(void* const* d_in, const int* in_sizes, int n_in,
                              void* d_out, int out_size, void* d_ws, size_t ws_size,
                              hipStream_t stream) {
  const float* ori_feature   = (const float*)d_in[0];
  const float* query_feature = (const float*)d_in[1];
  const int*   ori_indices   = (const int*)d_in[2];
  const int*   query_indices = (const int*)d_in[3];
  const float* pos_table     = (const float*)d_in[4];
  const float* in_proj_w     = (const float*)d_in[5];   // [384,128]
  const float* in_proj_b     = (const float*)d_in[6];   // [384]
  const float* out_w         = (const float*)d_in[7];   // [128,128]
  const float* out_b         = (const float*)d_in[8];   // [128]
  float* out = (float*)d_out;

  float* ws    = (float*)d_ws;
  float* q_in  = ws;                          // 8192*128
  float* kv_in = q_in  + (size_t)NQ * EMBED;  // 16384*128
  float* Qh    = kv_in + (size_t)NK * EMBED;  // 8192*128
  float* Kh    = Qh    + (size_t)NQ * EMBED;  // 16384*128
  float* Vh    = Kh    + (size_t)NK * EMBED;  // 16384*128
  float* AO    = Vh    + (size_t)NK * EMBED;  // 8192*128

  // 1) positional gather + add
  pos_add_kernel<<<NK, 128, 0, stream>>>(ori_feature,  ori_indices,  pos_table, kv_in);
  pos_add_kernel<<<NQ, 128, 0, stream>>>(query_feature, query_indices, pos_table, q_in);

  // 2) QKV projections (scale 1/sqrt(hd)=0.25 folded into Q)
  qkv_proj_kernel<LQ><<<dim3(EMBED / 16, NQ / 128), 256, 0, stream>>>(
      q_in, in_proj_w, in_proj_b, Qh, 0.25f);
  qkv_proj_kernel<LK><<<dim3(EMBED / 16, NK / 128), 256, 0, stream>>>(
      kv_in, in_proj_w + 128 * EMBED, in_proj_b + 128, Kh, 1.0f);
  qkv_proj_kernel<LK><<<dim3(EMBED / 16, NK / 128), 256, 0, stream>>>(
      kv_in, in_proj_w + 256 * EMBED, in_proj_b + 256, Vh, 1.0f);

  // 3) attention
  attn_kernel<<<dim3(LQ / 128, BATCH * HEADS), 256, 0, stream>>>(Qh, Kh, Vh, AO);

  // 4) output projection
  out_proj_kernel<<<dim3(EMBED / 16, NQ / 128), 256, 0, stream>>>(AO, out_w, out_b, out);
}